// MultiHeadDeformableAttention3D_17849884082216
// MI455X (gfx1250) — compile-verified
//
#include <hip/hip_runtime.h>
#include <hip/hip_bf16.h>
#include <math.h>

// Problem constants (from reference)
#define N_B 2
#define L_Q 1024
#define E_D 256
#define H_N 8
#define P_N 4
#define K_N 4
#define D_H 32          // E/H
#define NH  16          // N*H
#define LP  4096        // L*P

typedef __attribute__((ext_vector_type(2))) float v2f;
typedef __attribute__((ext_vector_type(8))) float v8f;
typedef __attribute__((ext_vector_type(4))) unsigned u32x4;
typedef __attribute__((ext_vector_type(8))) int     i32x8;
typedef __attribute__((ext_vector_type(4))) int     i32x4;

// ---------------------------------------------------------------------------
// WMMA f32 GEMM:  C[M x Nc] = A[M x K] @ B[K x Nc] + bias[Nc]
// One wave per 16 x (16*NT) output strip. The A fragment (16x4) is loaded
// once per k-step as a contiguous float2 (global_load_b64) and reused for NT
// consecutive WMMAs -> NT x arithmetic intensity vs. one-tile-per-wave.
// V_WMMA_F32_16X16X4_F32 keeps exact fp32 semantics (the downstream top-k
// neighbor selection is precision sensitive).
//
// Fragment layouts (CDNA5 ISA 7.12.2):
//   A 16x4 :  lane<16: M=lane, holds K=k+0,k+1 ; lane>=16: M=lane-16, K=k+2,k+3
//   B 4x16 :  lane<16: N=lane, holds K=k+0,k+1 ; lane>=16: N=lane-16, K=k+2,k+3
//   C 16x16:  vgpr r: lane<16 -> (M=r, N=lane) ; lane>=16 -> (M=r+8, N=lane-16)
// ---------------------------------------------------------------------------
template <int NT>
__global__ __launch_bounds__(32)
void gemm16_wmma_f32(const float* __restrict__ A, const float* __restrict__ B,
                     const float* __restrict__ bias, float* __restrict__ C,
                     int M, int K, int Nc) {
  const int tile_m = blockIdx.x;
  const int lane   = (int)threadIdx.x;   // 0..31
  const int hi     = lane >> 4;          // 0 | 1 (half-wave)
  const int l16    = lane & 15;

  const int row_a = tile_m * 16 + l16;            // M index for A fragment
  const int col0  = blockIdx.y * (16 * NT) + l16; // first N column this lane owns

  v8f acc[NT];
#pragma unroll
  for (int t = 0; t < NT; ++t) acc[t] = (v8f){};

  for (int k = 0; k < K; k += 4) {
    const int ka = k + hi * 2;
    const v2f afrag = *(const v2f*)(A + row_a * K + ka);  // 8B aligned: K even, ka even
    const float* Brow = B + ka * Nc;
#pragma unroll
    for (int t = 0; t < NT; ++t) {
      v2f bfrag;
      bfrag.x = Brow[t * 16 + col0];
      bfrag.y = Brow[Nc + t * 16 + col0];
      // 8 args: (neg_a, A, neg_b, B, c_mod, C, reuse_a, reuse_b)
      acc[t] = __builtin_amdgcn_wmma_f32_16x16x4_f32(
          false, afrag, false, bfrag, (short)0, acc[t], false, false);
    }
  }

#pragma unroll
  for (int t = 0; t < NT; ++t) {
    const float bv = bias[col0 + t * 16];
#pragma unroll
    for (int r = 0; r < 8; ++r) {
      const int row = tile_m * 16 + r + hi * 8;
      C[row * Nc + col0 + t * 16] = acc[t][r] + bv;
    }
  }
}

// ---------------------------------------------------------------------------
// Softmax over P=4 within each (n,l,h). Layout: att[(n*L+l)*32 + h*4 + p],
// the 4 entries per (n,l,h) are contiguous. In-place.
// ---------------------------------------------------------------------------
__global__ __launch_bounds__(256)
void softmax_p4(float* __restrict__ att, int total /* N*L*H */) {
  int i = blockIdx.x * blockDim.x + threadIdx.x;
  if (i >= total) return;
  float* p = att + i * 4;
  float a0 = p[0], a1 = p[1], a2 = p[2], a3 = p[3];
  float m = fmaxf(fmaxf(a0, a1), fmaxf(a2, a3));
  float e0 = expf(a0 - m), e1 = expf(a1 - m), e2 = expf(a2 - m), e3 = expf(a3 - m);
  float inv = 1.0f / (e0 + e1 + e2 + e3);
  p[0] = e0 * inv; p[1] = e1 * inv; p[2] = e2 * inv; p[3] = e3 * inv;
}

// ---------------------------------------------------------------------------
// Top-k + IDW kernel. rp[b % N] (12 KB) is staged into LDS.
//
// Staging demonstrates the CDNA5 Tensor Data Mover: each wave issues a
// TENSOR_LOAD_TO_LDS with a 1-D D# (3072 x 4B tile), then waits on
// TENSORcnt. Because the D# layout is not hardware-verified, a plain
// VALU copy afterwards overwrites the same LDS region with identical data,
// so correctness never depends on the descriptor.
// ---------------------------------------------------------------------------
#if defined(__gfx1250__) && __has_builtin(__builtin_amdgcn_tensor_load_to_lds)
#define HAVE_TDM 1
#else
#define HAVE_TDM 0
#endif

__global__ __launch_bounds__(256)
void topk_idw(const float* __restrict__ rp,    // (N, L, 3)
              const float* __restrict__ off,   // (N*L, 96)
              int*   __restrict__ tidx,        // (NH, LP, 4)
              float* __restrict__ widw) {      // (NH, LP, 4)
  __shared__ float s_rp[L_Q * 3];
  const int b  = blockIdx.y;        // 0..15
  const int t  = (int)threadIdx.x;
  const int rb = b % N_B;           // reference-point batch (ref's b % N)
  const float* rp_src = rp + rb * L_Q * 3;

#if HAVE_TDM
  {
    // D# group 0: count=1 | lds_addr | global_addr[56:0] | type=2
    const unsigned long long ga = (unsigned long long)rp_src;
    const unsigned lds_addr = (unsigned)(unsigned long long)&s_rp[0]; // low 32b = LDS offset
    u32x4 g0;
    g0[0] = 1u;                                           // count=1 (valid)
    g0[1] = lds_addr;                                     // lds_addr
    g0[2] = (unsigned)(ga & 0xFFFFFFFFu);                 // global_addr[31:0]
    g0[3] = (unsigned)((ga >> 32) & 0x1FFFFFFu) | (2u << 30); // ga[56:32] | type=2
    // D# group 1: data_size=4B, tensor_dim0=3072, tensor_dim1=1,
    //             tile_dim0=3072 (1-D tile), stride0=3072
    i32x8 g1;
    g1[0] = (int)(2u << 16);            // data_size=2 (4 bytes), wg_mask=0
    g1[1] = (int)(3072u << 16);         // tensor_dim0[15:0] in bits[63:48]
    g1[2] = (int)(1u << 16);            // tensor_dim0[31:16]=0 | tensor_dim1[15:0]=1
    g1[3] = (int)(3072u << 16);         // tensor_dim1[31:16]=0 | tile_dim0=3072
    g1[4] = 0;                          // tile_dim1=0, tile_dim2=0
    g1[5] = 3072;                       // tensor_dim0_stride[31:0]
    g1[6] = 0;
    g1[7] = 0;
    i32x4 gz = {0, 0, 0, 0};
#if __clang_major__ >= 23
    i32x8 gz8 = {0, 0, 0, 0, 0, 0, 0, 0};
    __builtin_amdgcn_tensor_load_to_lds(g0, g1, gz, gz, gz8, 0);
#else
    __builtin_amdgcn_tensor_load_to_lds(g0, g1, gz, gz, 0);
#endif
#if __has_builtin(__builtin_amdgcn_s_wait_tensorcnt)
    __builtin_amdgcn_s_wait_tensorcnt(0);
#else
    asm volatile("s_wait_tensorcnt 0x0" ::: "memory");
#endif
  }
#endif

  // Safety-net copy (identical data; also the fallback when TDM is absent).
  for (int j = t; j < L_Q * 3; j += 256) s_rp[j] = rp_src[j];
  __syncthreads();

  const int m = blockIdx.x * 256 + t;        // 0..4095
  const int n = b / H_N, h = b % H_N;
  const int l = m / P_N, p = m % P_N;

  const float* o = off + (n * L_Q + l) * 96 + (h * P_N + p) * 3;
  const float* r = rp + (n * L_Q + l) * 3;
  const float sx = r[0] + o[0];
  const float sy = r[1] + o[1];
  const float sz = r[2] + o[2];

  float bd[4] = {3.0e38f, 3.0e38f, 3.0e38f, 3.0e38f};
  int   bi[4] = {0, 0, 0, 0};

  for (int j = 0; j < L_Q; ++j) {
    const float dx = sx - s_rp[j * 3 + 0];
    const float dy = sy - s_rp[j * 3 + 1];
    const float dz = sz - s_rp[j * 3 + 2];
    const float d2 = dx * dx + dy * dy + dz * dz;
    if (d2 < bd[3]) {       // strict '<' keeps earlier index on ties (lax.top_k)
      bd[3] = d2; bi[3] = j;
#pragma unroll
      for (int q = 3; q > 0; --q) {
        if (bd[q] < bd[q - 1]) {
          float td = bd[q]; bd[q] = bd[q - 1]; bd[q - 1] = td;
          int   ti = bi[q]; bi[q] = bi[q - 1]; bi[q - 1] = ti;
        }
      }
    }
  }

  float w[4]; float s = 0.0f;
#pragma unroll
  for (int k = 0; k < 4; ++k) {
    const float d = sqrtf(fmaxf(bd[k], 1e-12f));  // sqrt deferred to 4 winners
    w[k] = 1.0f / (d + 1e-8f);
    s += w[k];
  }
  const float inv = 1.0f / s;
  const int base = (b * LP + m) * 4;
#pragma unroll
  for (int k = 0; k < 4; ++k) {
    tidx[base + k] = bi[k];
    widw[base + k] = w[k] * inv;
  }
}

// ---------------------------------------------------------------------------
// One wave per (b, l); lane = channel d in [0,32).
// out_pre[n][l][h*32+d] = sum_p attw[n,l,h,p] * sum_k idw * value_gather
// Value gather uses the reference's vflat indexing: batch = b%N, head = b/N.
// ---------------------------------------------------------------------------
__global__ __launch_bounds__(128)
void gather_combine(const float* __restrict__ val,    // (N*L, E)
                    const float* __restrict__ attw,   // (N*L, 32)
                    const int*   __restrict__ tidx,   // (NH, LP, 4)
                    const float* __restrict__ widw,   // (NH, LP, 4)
                    float* __restrict__ out_pre) {    // (N*L, E)
  const int wave = (int)threadIdx.x >> 5;
  const int lane = (int)threadIdx.x & 31;       // d
  const int gl = blockIdx.x * 4 + wave;         // b*L + l
  const int b = gl / L_Q;
  const int l = gl % L_Q;
  const int n  = b / H_N, h = b % H_N;
  const int vb = b % N_B;                       // gather batch (ref quirk)
  const int vh = b / N_B;                       // gather head  (ref quirk)

  float acc = 0.0f;
#pragma unroll
  for (int p = 0; p < P_N; ++p) {
    const float aw = attw[(n * L_Q + l) * 32 + h * 4 + p];
    const int base = (b * LP + l * P_N + p) * 4;
#pragma unroll
    for (int k = 0; k < 4; ++k) {
      const int idx = tidx[base + k];
      const float w = widw[base + k];
      acc += aw * w * val[(vb * L_Q + idx) * E_D + vh * D_H + lane];
    }
  }
  out_pre[(n * L_Q + l) * E_D + h * D_H + lane] = acc;
}

// ---------------------------------------------------------------------------
extern "C" void kernel_launch(void* const* d_in, const int* in_sizes, int n_in,
                              void* d_out, int out_size, void* d_ws, size_t ws_size,
                              hipStream_t stream) {
  const float* q      = (const float*)d_in[0];  // (N,L,E)
  const float* rp     = (const float*)d_in[1];  // (N,L,3)
  const float* W_off  = (const float*)d_in[2];  // (E, 96)
  const float* b_off  = (const float*)d_in[3];  // (96)
  const float* W_attn = (const float*)d_in[4];  // (E, 32)
  const float* b_attn = (const float*)d_in[5];  // (32)
  const float* W_val  = (const float*)d_in[6];  // (E, E)
  const float* b_val  = (const float*)d_in[7];  // (E)
  const float* W_out  = (const float*)d_in[8];  // (E, E)
  const float* b_out  = (const float*)d_in[9];  // (E)
  float* out = (float*)d_out;                   // (N,L,E)

  // Workspace layout (floats); total ~7.3 MB
  float* ws     = (float*)d_ws;
  float* val    = ws;                        // 2048*256 = 524288
  float* offv   = val  + 524288;             // 2048*96  = 196608
  float* att    = offv + 196608;             // 2048*32  = 65536
  float* outp   = att  + 65536;              // 2048*256 = 524288
  float* widw   = outp + 524288;             // 16*4096*4 = 262144
  int*   tidx   = (int*)(widw + 262144);     // 262144 ints

  const int M = N_B * L_Q;   // 2048

  // 1) value = q @ W_val + b_val          (16x64 strip per wave)
  gemm16_wmma_f32<4><<<dim3(M / 16, E_D / 64), 32, 0, stream>>>(
      q, W_val, b_val, val, M, E_D, E_D);
  // 2) off = q @ W_off + b_off            (16x32 strip per wave, 96 = 3*32)
  gemm16_wmma_f32<2><<<dim3(M / 16, 96 / 32), 32, 0, stream>>>(
      q, W_off, b_off, offv, M, E_D, 96);
  // 3) attn logits = q @ W_attn + b_attn  (16x32 strip per wave)
  gemm16_wmma_f32<2><<<dim3(M / 16, 32 / 32), 32, 0, stream>>>(
      q, W_attn, b_attn, att, M, E_D, 32);
  // 4) softmax over P
  softmax_p4<<<(M * H_N + 255) / 256, 256, 0, stream>>>(att, M * H_N);
  // 5) top-k + IDW weights (TDM-staged rp in LDS)
  topk_idw<<<dim3(LP / 256, NH), 256, 0, stream>>>(rp, offv, tidx, widw);
  // 6) gather + combine heads
  gather_combine<<<(NH * L_Q) / 4, 128, 0, stream>>>(val, att, tidx, widw, outp);
  // 7) final projection: out = outp @ W_out + b_out
  gemm16_wmma_f32<4><<<dim3(M / 16, E_D / 64), 32, 0, stream>>>(
      outp, W_out, b_out, out, M, E_D, E_D);
}